// PositionPredictor_49976239456311
// MI455X (gfx1250) — compile-verified
//
#include <hip/hip_runtime.h>

#define B_    2
#define L_    160
#define AD_   128
#define NAT_  37
#define NSC_  34
#define NS_   (L_ * NSC_)   // 5440, divisible by 16 and 4
#define HID_  512
#define EPS2_ 64.0f
#define INV2E2_ 0.0078125f  // 1/(2*EPS*EPS) = 1/128

typedef float v2f __attribute__((ext_vector_type(2)));
typedef float v8f __attribute__((ext_vector_type(8)));

__device__ __forceinline__ v8f wmma_f32x4(v2f a, v2f b, v8f c) {
#if defined(__gfx1250__) && __has_builtin(__builtin_amdgcn_wmma_f32_16x16x4_f32)
  // D = A(16x4 f32) * B(4x16 f32) + C(16x16 f32), wave32
  return __builtin_amdgcn_wmma_f32_16x16x4_f32(false, a, false, b, (short)0, c,
                                               false, false);
#else
  // Should not be reached on gfx1250; keeps host pass / probe compiling.
  c[0] += a.x * b.x + a.y * b.y;
  return c;
#endif
}

// ---------------------------------------------------------------------------
// Kernel 1: per-residue prep.
//   - argmax over 20 aa logits (first-max ties, matching jnp.argmax)
//   - atom_mask -> d_out[0 : B*L*37]
//   - px[b][n] = float4(x, y, z, valid ? |x|^2 : 1e30)   (validity folded)
//   - hf[b][n][d] = (scalar[d] + emb[d]*am) * am          (masked features)
// ---------------------------------------------------------------------------
__global__ __launch_bounds__(128)
void prep_kernel(const float* __restrict__ bb_pred,
                 const float* __restrict__ scalar_features,
                 const float* __restrict__ aa_pred,
                 const float* __restrict__ mask,
                 const float* __restrict__ pmask,
                 const float* __restrict__ atom_emb,
                 const float* __restrict__ pos_noise,
                 float* __restrict__ out_atom_mask,
                 float4* __restrict__ px,
                 float* __restrict__ hf)
{
  const int bl = blockIdx.x;          // b*L + l
  const int b  = bl / L_;
  const int l  = bl % L_;
  const int t  = threadIdx.x;         // 0..127
  const float mk = mask[bl];

  const float* aa = aa_pred + (size_t)bl * 20;
  int r = 0; float best = aa[0];
  #pragma unroll
  for (int i = 1; i < 20; ++i) { float v = aa[i]; if (v > best) { best = v; r = i; } }

  if (t < NAT_) out_atom_mask[(size_t)bl * NAT_ + t] = pmask[r * NAT_ + t] * mk;

  const float cax = bb_pred[(size_t)bl * 9 + 3 + 0] * mk;
  const float cay = bb_pred[(size_t)bl * 9 + 3 + 1] * mk;
  const float caz = bb_pred[(size_t)bl * 9 + 3 + 2] * mk;

  if (t < NSC_) {
    const float am = pmask[r * NAT_ + 3 + t] * mk;
    const int n = l * NSC_ + t;
    const float* pn = pos_noise + ((size_t)bl * NSC_ + t) * 3;
    const float x = cax + pn[0], y = cay + pn[1], z = caz + pn[2];
    const float x2 = x * x + y * y + z * z;
    px[(size_t)b * NS_ + n] = make_float4(x, y, z, am > 0.0f ? x2 : 1.0e30f);
  }

  const float sd = scalar_features[(size_t)bl * AD_ + t];
  float* hfrow = hf + ((size_t)b * NS_ + (size_t)l * NSC_) * AD_ + t;
  for (int s = 0; s < NSC_; ++s) {
    const float am = pmask[r * NAT_ + 3 + s] * mk;
    hfrow[(size_t)s * AD_] = (sd + atom_emb[(3 + s) * AD_ + t] * am) * am;
  }
}

// ---------------------------------------------------------------------------
// Kernel 2: fused distance-weighted aggregation (fp32 WMMA GEMM with A built
// in-register from distances) + gate MLP (fp32 WMMA) + position update.
// One wave (32 threads) per 16-row tile; grid = (NS/16, B).
// ---------------------------------------------------------------------------
__global__ __launch_bounds__(32)
void fused_kernel(const float4* __restrict__ px,
                  const float*  __restrict__ hf,
                  const float*  __restrict__ W1,
                  const float*  __restrict__ b1,
                  const float*  __restrict__ W2,
                  const float*  __restrict__ b2,
                  float* __restrict__ out_pos)
{
  __shared__ float ldsP[16 * 4];      // per-row: wx, wy, wz, deg
  __shared__ float ldsH[16 * AD_];    // inv-scaled h_agg tile (8 KB)
  __shared__ float ldsO[16 * 3];      // reduced pre-tanh MLP output

  const int b       = blockIdx.y;
  const int rowBase = blockIdx.x * 16;
  const int lane    = threadIdx.x;
  const int cn      = lane & 15;      // A-row / B-col / C-col index
  const int hi      = lane >> 4;
  const int klo     = hi * 2;         // K indices this lane owns: klo, klo+1

  const float4* pxb = px + (size_t)b * NS_;
  const float*  hfb = hf + (size_t)b * NS_ * AD_;

  const float4 pn = pxb[rowBase + cn];  // this lane's A-row position data

  v8f accH[8] = {};                   // 16x128 h_agg accumulator (8 col tiles)
  v8f accP    = {};                   // 16x16, cols 0..3 = w@[x,y,z,1]

  for (int mb = 0; mb < NS_; mb += 4) {
    const float4 pm0 = pxb[mb + klo];
    const float4 pm1 = pxb[mb + klo + 1];
    const float d20 = pn.w + pm0.w - 2.0f * (pn.x*pm0.x + pn.y*pm0.y + pn.z*pm0.z);
    const float d21 = pn.w + pm1.w - 2.0f * (pn.x*pm1.x + pn.y*pm1.y + pn.z*pm1.z);
    v2f aw;
    aw.x = (d20 < EPS2_) ? __expf(-d20 * INV2E2_) : 0.0f;
    aw.y = (d21 < EPS2_) ? __expf(-d21 * INV2E2_) : 0.0f;

    const float* h0 = hfb + (size_t)(mb + klo) * AD_ + cn;
    #pragma unroll
    for (int ft = 0; ft < 8; ++ft) {
      v2f bf;
      bf.x = h0[ft * 16];
      bf.y = h0[AD_ + ft * 16];
      accH[ft] = wmma_f32x4(aw, bf, accH[ft]);
    }
    v2f bp;
    bp.x = (cn == 0) ? pm0.x : (cn == 1) ? pm0.y : (cn == 2) ? pm0.z
                                                 : (cn == 3) ? 1.0f : 0.0f;
    bp.y = (cn == 0) ? pm1.x : (cn == 1) ? pm1.y : (cn == 2) ? pm1.z
                                                 : (cn == 3) ? 1.0f : 0.0f;
    accP = wmma_f32x4(aw, bp, accP);
  }

  // Stage (w@x, deg) through LDS so every lane can see its row's scalars.
  if (cn < 4) {
    #pragma unroll
    for (int j = 0; j < 8; ++j) ldsP[(j + 8 * hi) * 4 + cn] = accP[j];
  }
  __syncthreads();

  float invRow[8];
  #pragma unroll
  for (int j = 0; j < 8; ++j)
    invRow[j] = 1.0f / (ldsP[(j + 8 * hi) * 4 + 3] + 1.0f);

  // h_agg * inv -> LDS (C-fragment layout to row-major re-layout for next GEMM)
  #pragma unroll
  for (int ft = 0; ft < 8; ++ft) {
    #pragma unroll
    for (int j = 0; j < 8; ++j)
      ldsH[(j + 8 * hi) * AD_ + ft * 16 + cn] = accH[ft][j] * invRow[j];
  }
  __syncthreads();

  // MLP: hidden = relu(z @ W1 + b1); out3 = hidden @ W2. z = [hf | h_agg].
  float pout0[8] = {}, pout1[8] = {}, pout2[8] = {};
  for (int ct = 0; ct < 32; ++ct) {
    const int c0 = ct * 16;
    v8f acc = {};
    for (int kb = 0; kb < 2 * AD_; kb += 4) {
      const int k0 = kb + klo;
      v2f az, bw;
      az.x = (k0     < AD_) ? hfb[(size_t)(rowBase + cn) * AD_ + k0]
                            : ldsH[cn * AD_ + (k0 - AD_)];
      az.y = (k0 + 1 < AD_) ? hfb[(size_t)(rowBase + cn) * AD_ + k0 + 1]
                            : ldsH[cn * AD_ + (k0 + 1 - AD_)];
      bw.x = W1[(size_t)k0 * HID_ + c0 + cn];
      bw.y = W1[(size_t)(k0 + 1) * HID_ + c0 + cn];
      acc = wmma_f32x4(az, bw, acc);
    }
    const float bb  = b1[c0 + cn];
    const float w20 = W2[(c0 + cn) * 3 + 0];
    const float w21 = W2[(c0 + cn) * 3 + 1];
    const float w22 = W2[(c0 + cn) * 3 + 2];
    #pragma unroll
    for (int j = 0; j < 8; ++j) {
      float hv = fmaxf(acc[j] + bb, 0.0f);
      pout0[j] += hv * w20;
      pout1[j] += hv * w21;
      pout2[j] += hv * w22;
    }
  }

  // Deterministic shuffle reduction over the 16 column lanes of each half.
  #pragma unroll
  for (int j = 0; j < 8; ++j) {
    #pragma unroll
    for (int m = 1; m <= 8; m <<= 1) {
      pout0[j] += __shfl_xor(pout0[j], m, 32);
      pout1[j] += __shfl_xor(pout1[j], m, 32);
      pout2[j] += __shfl_xor(pout2[j], m, 32);
    }
  }
  if (cn == 0) {
    #pragma unroll
    for (int j = 0; j < 8; ++j) {
      ldsO[(j + 8 * hi) * 3 + 0] = pout0[j];
      ldsO[(j + 8 * hi) * 3 + 1] = pout1[j];
      ldsO[(j + 8 * hi) * 3 + 2] = pout2[j];
    }
  }
  __syncthreads();

  // Final: gate = tanh(out3 + b2); disp = agg_pos * gate * valid; write pos.
  if (lane < 16) {
    const int row = lane;
    const int n   = rowBase + row;
    const float4 pr = pxb[n];
    const float valid = (pr.w < 1.0e29f) ? 1.0f : 0.0f;
    const float wx = ldsP[row * 4 + 0], wy = ldsP[row * 4 + 1];
    const float wz = ldsP[row * 4 + 2], dg = ldsP[row * 4 + 3];
    const float inv = 1.0f / (dg + 1.0f);
    const float gx = tanhf(ldsO[row * 3 + 0] + b2[0]);
    const float gy = tanhf(ldsO[row * 3 + 1] + b2[1]);
    const float gz = tanhf(ldsO[row * 3 + 2] + b2[2]);
    float* o = out_pos + ((size_t)b * NS_ + n) * 3;
    o[0] = pr.x + ((pr.x * dg - wx) * inv) * gx * valid;
    o[1] = pr.y + ((pr.y * dg - wy) * inv) * gy * valid;
    o[2] = pr.z + ((pr.z * dg - wz) * inv) * gz * valid;
  }
}

// ---------------------------------------------------------------------------
extern "C" void kernel_launch(void* const* d_in, const int* in_sizes, int n_in,
                              void* d_out, int out_size, void* d_ws, size_t ws_size,
                              hipStream_t stream)
{
  (void)in_sizes; (void)n_in; (void)out_size; (void)ws_size;
  const float* bb_pred   = (const float*)d_in[0];
  const float* scalarf   = (const float*)d_in[1];
  const float* aa_pred   = (const float*)d_in[2];
  // d_in[3] residue_batch: unused by the reference computation
  const float* mask      = (const float*)d_in[4];
  const float* pmask     = (const float*)d_in[5];
  const float* atom_emb  = (const float*)d_in[6];
  const float* W1        = (const float*)d_in[7];
  const float* b1        = (const float*)d_in[8];
  const float* W2        = (const float*)d_in[9];
  const float* b2        = (const float*)d_in[10];
  const float* pos_noise = (const float*)d_in[11];

  float* out = (float*)d_out;                   // [0 : B*L*37] atom_mask,
                                                // [B*L*37 : ] updated_pos
  float* px = (float*)d_ws;                     // B*NS*4 floats (float4)
  float* hf = px + (size_t)B_ * NS_ * 4;        // B*NS*128 floats (~5.6 MB)

  prep_kernel<<<B_ * L_, 128, 0, stream>>>(
      bb_pred, scalarf, aa_pred, mask, pmask, atom_emb, pos_noise,
      out, (float4*)px, hf);

  dim3 grid(NS_ / 16, B_);
  fused_kernel<<<grid, 32, 0, stream>>>(
      (const float4*)px, hf, W1, b1, W2, b2, out + (size_t)B_ * L_ * NAT_);
}